// UnsupervisedWrapper_64261300682946
// MI455X (gfx1250) — compile-verified
//
#include <hip/hip_runtime.h>

typedef __attribute__((ext_vector_type(2))) float v2f;
typedef __attribute__((ext_vector_type(8))) float v8f;

#define MARGIN_F  0.5f
#define EPS_F     1e-5f
#define PD_EPS_F  1e-6f

// Problem geometry is fixed by the reference (8192 x 256). Making the inner
// dimension a compile-time constant turns every operand address into
// base + immediate (24-bit IOFFSET), eliminating per-iteration 64-bit adds.
constexpr int D = 256;

// Column tiles per wave (register blocking): one A fetch feeds NT WMMAs.
#define NT 4

// ---------------------------------------------------------------------------
// Register-blocked, software-pipelined Gram tiles: one wave computes a
// 16x(16*NT) block of A(rows) x B(rows)^T with V_WMMA_F32_16X16X4_F32.
//
// Operand layouts (ISA 7.12.2, fp32):
//   A 16x4 : lane<16 -> M=lane,    K={0,1} in v0/v1; lane>=16 -> M=lane-16, K={2,3}
//   B 4x16 : mirrored on N; for B = k^T the per-lane indexing into the
//            row-major source is identical to A's.
//   C      : VGPR r = (M=r, N=lane) lanes 0-15, (M=r+8, N=lane-16) lanes 16-31.
//
// Pipeline: loads for k-step kk+4 are issued before the WMMAs of k-step kk,
// so the s_wait before each WMMA group only covers loads that have had a full
// iteration (4 WMMAs + clause issue) to complete.
// ---------------------------------------------------------------------------
__device__ __forceinline__ void wmma_gram_tiles(const float* __restrict__ A,
                                                const float* __restrict__ B,
                                                int row_base, int col_base,
                                                v8f c[NT]) {
  const int lane = threadIdx.x & 31;
  const int half = lane >> 4;   // 0 or 1
  const int l16  = lane & 15;
  const float* __restrict__ arow = A + (size_t)(row_base + l16) * D + 2 * half;
  const float* __restrict__ brow = B + (size_t)(col_base + l16) * D + 2 * half;

  // Prologue: k-step 0.
  v2f a_c = *(const v2f*)arow;
  v2f b_c[NT];
#pragma unroll
  for (int t = 0; t < NT; ++t) b_c[t] = *(const v2f*)(brow + t * 16 * D);

#pragma unroll 8
  for (int kk = 0; kk < D - 4; kk += 4) {
    // Issue next k-step's loads first (constant offsets off shared bases).
    v2f a_n = *(const v2f*)(arow + kk + 4);
    v2f b_n[NT];
#pragma unroll
    for (int t = 0; t < NT; ++t)
      b_n[t] = *(const v2f*)(brow + t * 16 * D + kk + 4);

    // Compute current k-step.
#pragma unroll
    for (int t = 0; t < NT; ++t)
      c[t] = __builtin_amdgcn_wmma_f32_16x16x4_f32(false, a_c, false, b_c[t],
                                                   (short)0, c[t], false, false);
    // Rotate.
    a_c = a_n;
#pragma unroll
    for (int t = 0; t < NT; ++t) b_c[t] = b_n[t];
  }

  // Epilogue: last k-step.
#pragma unroll
  for (int t = 0; t < NT; ++t)
    c[t] = __builtin_amdgcn_wmma_f32_16x16x4_f32(false, a_c, false, b_c[t],
                                                 (short)0, c[t], false, false);
}

__device__ __forceinline__ float wave_sum32(float s) {
#pragma unroll
  for (int off = 16; off; off >>= 1) s += __shfl_xor(s, off, 32);
  return s;
}

// ---------------------------------------------------------------------------
// Kernel 1: fused margin loss over sim = q k^T (never materialized).
// Grid: (n/16 row tiles, n/512 col groups); 8 waves/WG, each wave a 16x64
// macro-tile. One deterministic partial per workgroup.
// ---------------------------------------------------------------------------
__global__ void __launch_bounds__(256)
loss_kernel(const float* __restrict__ q, const float* __restrict__ k,
            float* __restrict__ partial, int n) {
  const int wave     = threadIdx.x >> 5;
  const int row_base = blockIdx.x * 16;
  const int col_base = (blockIdx.y * 8 + wave) * (16 * NT);

  v8f c[NT];
#pragma unroll
  for (int t = 0; t < NT; ++t) c[t] = (v8f){};
  wmma_gram_tiles(q, k, row_base, col_base, c);

  float s = 0.0f;
#pragma unroll
  for (int t = 0; t < NT; ++t) {
#pragma unroll
    for (int r = 0; r < 8; ++r) {
      const float v = c[t][r];
      if (v < 1.0f - EPS_F) s += 1.0f - v;   // pos_pair contribution
      if (v > MARGIN_F)     s += v;          // neg_pair contribution
    }
  }
  s = wave_sum32(s);

  __shared__ float ws8[8];
  if ((threadIdx.x & 31) == 0) ws8[wave] = s;
  __syncthreads();
  if (threadIdx.x == 0) {
    float t = 0.0f;
#pragma unroll
    for (int w = 0; w < 8; ++w) t += ws8[w];
    partial[(size_t)blockIdx.y * gridDim.x + blockIdx.x] = t;
  }
}

// ---------------------------------------------------------------------------
// Kernel 2: per-row argmax of dots = q q^T, diagonal forced to -1.0.
// One WG per 16-row stripe; wave w sweeps 64-column super-tiles w, w+8, ...
// Running (max, idx) kept in the accumulator layout; ties -> smallest column
// index (== jnp.argmax first-occurrence). Next super-tile is hint-prefetched
// (global_prefetch_b8) before the K-loop.
// ---------------------------------------------------------------------------
__global__ void __launch_bounds__(256)
argmax_kernel(const float* __restrict__ q, int* __restrict__ I, int n) {
  const int wave     = threadIdx.x >> 5;
  const int lane     = threadIdx.x & 31;
  const int half     = lane >> 4;
  const int l16      = lane & 15;
  const int row_base = blockIdx.x * 16;
  const int superTiles = n / (16 * NT);     // 64-column super-tiles

  float rmax[8];
  int   ridx[8];
#pragma unroll
  for (int r = 0; r < 8; ++r) { rmax[r] = -1e30f; ridx[r] = 0; }

  for (int st = wave; st < superTiles; st += 8) {
    const int col_base = st * (16 * NT);

    const int nst = st + 8;
    if (nst < superTiles) {
#pragma unroll
      for (int t = 0; t < NT; ++t)
        __builtin_prefetch(q + (size_t)(nst * (16 * NT) + 16 * t + l16) * D, 0, 1);
    }

    v8f c[NT];
#pragma unroll
    for (int t = 0; t < NT; ++t) c[t] = (v8f){};
    wmma_gram_tiles(q, q, row_base, col_base, c);

#pragma unroll
    for (int t = 0; t < NT; ++t) {
#pragma unroll
      for (int r = 0; r < 8; ++r) {
        const int i = row_base + r + 8 * half;        // global row of this slot
        const int j = col_base + 16 * t + l16;        // global column
        const float v = (i == j) ? -1.0f : c[t][r];   // exclude self
        if (v > rmax[r]) { rmax[r] = v; ridx[r] = j; }
      }
    }
  }

  // Butterfly argmax across the 16 lanes of each half-wave.
#pragma unroll
  for (int off = 1; off < 16; off <<= 1) {
#pragma unroll
    for (int r = 0; r < 8; ++r) {
      const float om = __shfl_xor(rmax[r], off, 16);
      const int   oi = __shfl_xor(ridx[r], off, 16);
      if (om > rmax[r] || (om == rmax[r] && oi < ridx[r])) {
        rmax[r] = om; ridx[r] = oi;
      }
    }
  }

  __shared__ float smax[8][16];
  __shared__ int   sidx[8][16];
  if (l16 == 0) {
#pragma unroll
    for (int r = 0; r < 8; ++r) {
      const int m = r + 8 * half;
      smax[wave][m] = rmax[r];
      sidx[wave][m] = ridx[r];
    }
  }
  __syncthreads();

  if (threadIdx.x < 16) {
    float bm = -1e30f;
    int   bi = 0;
#pragma unroll
    for (int w = 0; w < 8; ++w) {
      const float v = smax[w][threadIdx.x];
      const int   j = sidx[w][threadIdx.x];
      if (v > bm || (v == bm && j < bi)) { bm = v; bi = j; }
    }
    I[row_base + threadIdx.x] = bi;
  }
}

// ---------------------------------------------------------------------------
// Kernel 3: -log ||q[i] - q[I[i]] + 1e-6||, one wave per row.
// ---------------------------------------------------------------------------
__global__ void __launch_bounds__(256)
dist_kernel(const float* __restrict__ q, const int* __restrict__ I,
            float* __restrict__ logd, int n) {
  const int wave = threadIdx.x >> 5;
  const int lane = threadIdx.x & 31;
  const int row  = blockIdx.x * 8 + wave;
  if (row >= n) return;

  const float* __restrict__ qi = q + (size_t)row * D;
  const float* __restrict__ qj = q + (size_t)I[row] * D;

  float s = 0.0f;
#pragma unroll
  for (int t = lane; t < D; t += 32) {
    const float dv = qi[t] - qj[t] + PD_EPS_F;
    s += dv * dv;
  }
  s = wave_sum32(s);
  if (lane == 0) logd[row] = -0.5f * __logf(s);  // -log(sqrt(s))
}

// ---------------------------------------------------------------------------
// Kernel 4: deterministic final reductions -> d_out[0]=loss, d_out[1]=reg.
// ---------------------------------------------------------------------------
__global__ void __launch_bounds__(256)
final_kernel(const float* __restrict__ lossPart, int nPart,
             const float* __restrict__ logd, int n, float* __restrict__ out) {
  __shared__ float red[256];

  float s = 0.0f;
  for (int i = threadIdx.x; i < nPart; i += 256) s += lossPart[i];
  red[threadIdx.x] = s;
  __syncthreads();
  for (int off = 128; off; off >>= 1) {
    if (threadIdx.x < off) red[threadIdx.x] += red[threadIdx.x + off];
    __syncthreads();
  }
  if (threadIdx.x == 0) out[0] = red[0] / (float)n;
  __syncthreads();

  s = 0.0f;
  for (int i = threadIdx.x; i < n; i += 256) s += logd[i];
  red[threadIdx.x] = s;
  __syncthreads();
  for (int off = 128; off; off >>= 1) {
    if (threadIdx.x < off) red[threadIdx.x] += red[threadIdx.x + off];
    __syncthreads();
  }
  if (threadIdx.x == 0) out[1] = red[0] / (float)n;
}

// ---------------------------------------------------------------------------
extern "C" void kernel_launch(void* const* d_in, const int* in_sizes, int n_in,
                              void* d_out, int out_size, void* d_ws, size_t ws_size,
                              hipStream_t stream) {
  const float* q = (const float*)d_in[0];
  const float* k = (const float*)d_in[1];
  float* out = (float*)d_out;

  const int n = in_sizes[0] / D;            // 8192

  const int rowTiles  = n / 16;             // 512
  const int colGroups = n / (128 * NT);     // 16 (8 waves x 64 cols per WG)
  const int nPart     = rowTiles * colGroups;

  // d_ws layout: [nPart floats loss partials][n ints I][n floats logd]
  float* lossPart = (float*)d_ws;
  int*   Iidx     = (int*)((char*)d_ws + (size_t)nPart * sizeof(float));
  float* logd     = (float*)((char*)d_ws + (size_t)nPart * sizeof(float)
                                         + (size_t)n * sizeof(int));

  loss_kernel<<<dim3(rowTiles, colGroups), 256, 0, stream>>>(q, k, lossPart, n);
  argmax_kernel<<<rowTiles, 256, 0, stream>>>(q, Iidx, n);
  dist_kernel<<<n / 8, 256, 0, stream>>>(q, Iidx, logd, n);
  final_kernel<<<1, 256, 0, stream>>>(lossPart, nPart, logd, n, out);
}